// CC_attention_74500502716432
// MI455X (gfx1250) — compile-verified
//
#include <hip/hip_runtime.h>

typedef __attribute__((ext_vector_type(2))) float v2f;
typedef __attribute__((ext_vector_type(8))) float v8f;

#define B_    8
#define C_    256
#define H_    128
#define W_    128
#define KP_   16      // pooled dimension (128/8)
#define EROW  4096    // energy row length = H*kH + W*kW
#define LPAD  132     // pitch for [16][LPAD] tiles (132 % 64 == 4 -> conflict-free)
#define APAD  20      // pitch for [128][APAD] transposed tiles; 80B rows, 16B aligned
#define TSZ   2560    // floats per tile buffer (max(16*132, 128*20) = 2560)
#define CSPLIT 4      // 4-way split of the channel reduction in energy kernel
#define CCHUNK 64     // C_/CSPLIT

// ---- workspace layout (floats) ----
#define KH_OFF  0u
#define KW_OFF  4194304u
#define VH_OFF  8388608u
#define VW_OFF  12582912u
#define ENP_OFF 16777216u            // 4 * 32768 energy partials
#define EN_OFF  16908288u            // 32768 reduced energies
#define ATT_OFF 16941056u            // 32768 attention weights
#define MM_OFF  16973824u            // min, max

// ---------------------------------------------------------------------------
// CDNA5 async global->LDS copies (ASYNCcnt path, no VGPR round trip)
// ---------------------------------------------------------------------------
__device__ __forceinline__ void async_b128(unsigned int lds, const void* g) {
  asm volatile("global_load_async_to_lds_b128 %0, %1, off"
               :: "v"(lds), "v"((unsigned long long)(size_t)g) : "memory");
}
__device__ __forceinline__ unsigned int lds_off(const void* p) {
  return (unsigned int)(size_t)p;    // low 32 bits of generic ptr = LDS byte offset
}

// ---------------------------------------------------------------------------
// Kernel 1: average pooling along H (->keyH/valH) and along W (->keyW/valW)
// ---------------------------------------------------------------------------
__global__ __launch_bounds__(256) void pool_kernel(
    const float* __restrict__ t1, const float* __restrict__ t2,
    float* __restrict__ keyH, float* __restrict__ keyW,
    float* __restrict__ valH, float* __restrict__ valW) {
  int idx = blockIdx.x * 256 + threadIdx.x;   // 0 .. 4194303
  {
    int w = idx & 127, k = (idx >> 7) & 15, bc = idx >> 11;
    const float* p1 = t1 + ((size_t)bc * H_ + k * 8) * W_ + w;
    const float* p2 = t2 + ((size_t)bc * H_ + k * 8) * W_ + w;
    float s1 = 0.f, s2 = 0.f;
#pragma unroll
    for (int j = 0; j < 8; ++j) { s1 += p1[j * W_]; s2 += p2[j * W_]; }
    keyH[idx] = s1 * 0.125f;
    valH[idx] = s2 * 0.125f;
  }
  {
    int k = idx & 15, h = (idx >> 4) & 127, bc = idx >> 11;
    const float* p1 = t1 + ((size_t)bc * H_ + h) * W_ + k * 8;
    const float* p2 = t2 + ((size_t)bc * H_ + h) * W_ + k * 8;
    float s1 = 0.f, s2 = 0.f;
#pragma unroll
    for (int j = 0; j < 8; ++j) { s1 += p1[j]; s2 += p2[j]; }
    keyW[idx] = s1 * 0.125f;
    valW[idx] = s2 * 0.125f;
  }
}

// ---------------------------------------------------------------------------
// Kernel 2: energy GEMMs via V_WMMA_F32_16X16X4_F32
//   block = ((b*16 + t)*4 + s4); t<8 -> energy_H tile, else energy_W tile;
//   s4 selects a 64-channel slice of the K reduction (partials to ws).
//   3-buffer async pipeline, prefetch depth 2, uniform 4xB128 batches.
//   H branch tiles: [16][LPAD] (row-major); W branch tiles: [128][APAD]
//   (transposed so global reads stay 16B-contiguous). Fragment address is
//   rowOff + kk*kStride in both cases.
// ---------------------------------------------------------------------------
__global__ __launch_bounds__(256) void energy_kernel(
    const float* __restrict__ t2, const float* __restrict__ keyH,
    const float* __restrict__ keyW, float* __restrict__ epart) {
  __shared__ __align__(16) float smem[3][2][TSZ];   // 60KB: 3 bufs x (A,B) tiles

  int tid  = threadIdx.x;
  int wave = tid >> 5, lane = tid & 31;
  int blk = blockIdx.x;
  int s4 = blk & 3;
  int t  = (blk >> 2) & 15;
  int b  = blk >> 6;
  bool isH = (t < 8);
  int m0 = (t & 7) * 16;
  int c0 = s4 * CCHUNK;
  int lm = lane & 15;
  int koff = (lane >> 4) * 2;      // lanes 0-15: K 0,1 ; lanes 16-31: K 2,3

  auto issue = [&](int buf, int c) {   // 4 async B128 per thread per batch
    const float* tb = t2 + (size_t)(b * C_ + c) * (H_ * W_);
    unsigned int aB = lds_off(&smem[buf][0][0]);
    unsigned int bB = lds_off(&smem[buf][1][0]);
    if (isH) {
      const float* kb = keyH + (size_t)(b * C_ + c) * (KP_ * W_);
#pragma unroll
      for (int r = 0; r < 2; ++r) {
        int slot = tid + r * 256;                  // 512 float4 slots per tile
        int m = slot >> 5, w4 = (slot & 31) * 4;
        async_b128(aB + (unsigned)(m * LPAD + w4) * 4u, tb + (m0 + m) * W_ + w4);
        async_b128(bB + (unsigned)(m * LPAD + w4) * 4u, kb + m * W_ + w4);
      }
    } else {
      const float* kb = keyW + (size_t)(b * C_ + c) * (H_ * KP_);
#pragma unroll
      for (int r = 0; r < 2; ++r) {
        int slot = tid + r * 256;
        int h = slot >> 2, e4 = (slot & 3) * 4;
        async_b128(aB + (unsigned)(h * APAD + e4) * 4u, tb + h * W_ + m0 + e4);
        async_b128(bB + (unsigned)(h * APAD + e4) * 4u, kb + h * KP_ + e4);
      }
    }
  };

  issue(0, c0);
  issue(1, c0 + 1);

  int rowOff  = isH ? lm * LPAD : lm;   // block-uniform layout selectors
  int kStride = isH ? 1 : APAD;
  int kbase   = wave * 16;

  v8f acc = {};
  int bufC = 0, bufP = 2;
  for (int cc = 0; cc < CCHUNK - 1; ++cc) {
    asm volatile("s_wait_asynccnt 4" ::: "memory");  // batch cc landed
    __syncthreads();                                 // all waves' copies visible
    if (cc + 2 < CCHUNK) issue(bufP, c0 + cc + 2);
    const float* tA = &smem[bufC][0][0];
    const float* tB = &smem[bufC][1][0];
#pragma unroll
    for (int s = 0; s < 4; ++s) {
      int kk = kbase + s * 4 + koff;
      int o0 = rowOff + kk * kStride;
      v2f a  = { tA[o0], tA[o0 + kStride] };
      v2f bb = { tB[o0], tB[o0 + kStride] };
      acc = __builtin_amdgcn_wmma_f32_16x16x4_f32(false, a, false, bb,
                                                  (short)0, acc, false, false);
    }
    bufC = (bufC == 2) ? 0 : bufC + 1;
    bufP = (bufP == 2) ? 0 : bufP + 1;
  }
  // peeled final iteration: no newer batches -> must drain to 0
  {
    asm volatile("s_wait_asynccnt 0" ::: "memory");
    __syncthreads();
    const float* tA = &smem[bufC][0][0];
    const float* tB = &smem[bufC][1][0];
#pragma unroll
    for (int s = 0; s < 4; ++s) {
      int kk = kbase + s * 4 + koff;
      int o0 = rowOff + kk * kStride;
      v2f a  = { tA[o0], tA[o0 + kStride] };
      v2f bb = { tB[o0], tB[o0 + kStride] };
      acc = __builtin_amdgcn_wmma_f32_16x16x4_f32(false, a, false, bb,
                                                  (short)0, acc, false, false);
    }
  }
  // reduce the 8 wave partials (Cred aliases tile memory; safe after barrier)
  float* Cred = &smem[0][0][0];
  __syncthreads();
#pragma unroll
  for (int j = 0; j < 8; ++j)
    Cred[wave * 256 + (j + ((lane >> 4) << 3)) * 16 + lm] = acc[j];
  __syncthreads();
  int mr = tid >> 4, nc = tid & 15;
  float s = 0.f;
#pragma unroll
  for (int w8 = 0; w8 < 8; ++w8) s += Cred[w8 * 256 + mr * 16 + nc];
  epart[(size_t)s4 * (B_ * EROW) + (size_t)b * EROW + (isH ? 0 : 2048) +
        (m0 + mr) * 16 + nc] = s;
}

// ---------------------------------------------------------------------------
// Kernel 2b: deterministic fixed-order sum of the 4 channel-slice partials
// ---------------------------------------------------------------------------
__global__ __launch_bounds__(256) void reduce_energy_kernel(
    const float* __restrict__ part, float* __restrict__ e) {
  int i = blockIdx.x * 256 + threadIdx.x;   // 0 .. 32767
  float s = part[i];
#pragma unroll
  for (int s4 = 1; s4 < CSPLIT; ++s4) s += part[s4 * (B_ * EROW) + i];
  e[i] = s;
}

// ---------------------------------------------------------------------------
// Kernel 3: global min/max over all B*4096 energies (single block)
// ---------------------------------------------------------------------------
__global__ __launch_bounds__(256) void minmax_kernel(
    const float* __restrict__ e, float* __restrict__ mm) {
  __shared__ float smin[256], smax[256];
  int tid = threadIdx.x;
  float lmin = 3.4e38f, lmax = -3.4e38f;
  for (int i = tid; i < B_ * EROW; i += 256) {
    float v = e[i];
    lmin = fminf(lmin, v); lmax = fmaxf(lmax, v);
  }
  smin[tid] = lmin; smax[tid] = lmax;
  __syncthreads();
  for (int s = 128; s > 0; s >>= 1) {
    if (tid < s) {
      smin[tid] = fminf(smin[tid], smin[tid + s]);
      smax[tid] = fmaxf(smax[tid], smax[tid + s]);
    }
    __syncthreads();
  }
  if (tid == 0) { mm[0] = smin[0]; mm[1] = smax[0]; }
}

// ---------------------------------------------------------------------------
// Kernel 4: normalize + per-row softmax (one block per b)
// ---------------------------------------------------------------------------
__global__ __launch_bounds__(256) void softmax_kernel(
    const float* __restrict__ e, const float* __restrict__ mm,
    float* __restrict__ att) {
  __shared__ float red[256];
  __shared__ float invTot;
  int tid = threadIdx.x, b = blockIdx.x;
  float emin = mm[0];
  float inv  = 1.f / (mm[1] - mm[0]);
  const float* er = e + (size_t)b * EROW;
  float* ar = att + (size_t)b * EROW;
  float lsum = 0.f;
  for (int i = tid; i < EROW; i += 256) {
    float x = expf((er[i] - emin) * inv);   // inputs in [0,1]: no max-shift needed
    ar[i] = x; lsum += x;
  }
  red[tid] = lsum;
  __syncthreads();
  for (int s = 128; s > 0; s >>= 1) {
    if (tid < s) red[tid] += red[tid + s];
    __syncthreads();
  }
  if (tid == 0) invTot = 1.f / red[0];
  __syncthreads();
  float it = invTot;
  for (int i = tid; i < EROW; i += 256) ar[i] *= it;
}

// ---------------------------------------------------------------------------
// Kernel 5: out = 0.5*(attH@valH + valW@attW^T) + t2, one block per (b,c)
//   Both GEMMs: M=128, N=128, K=16 -> 8x8 tiles, 4+4 WMMA k-steps each.
//   Tiles staged with async B128 copies.
// ---------------------------------------------------------------------------
__global__ __launch_bounds__(256) void out_kernel(
    const float* __restrict__ t2, const float* __restrict__ valH,
    const float* __restrict__ valW, const float* __restrict__ att,
    float* __restrict__ out) {
  __shared__ __align__(16) float VHs[16][LPAD];    // VHs[k][w] = valH[b,c,k,w]
  __shared__ __align__(16) float AWs[128][APAD];   // AWs[w][k] = attW[b,w,k]
  int tid = threadIdx.x, wave = tid >> 5, lane = tid & 31;
  int bc = blockIdx.x;               // b*C + c
  int b = bc >> 8;                   // C_ == 256
  const float* vh = valH + (size_t)bc * (KP_ * W_);
  const float* aw = att + (size_t)b * EROW + 2048;    // flat w*16+k
  {
    unsigned int vB = lds_off(&VHs[0][0]);
    unsigned int aB = lds_off(&AWs[0][0]);
#pragma unroll
    for (int r = 0; r < 2; ++r) {
      int slot = tid + r * 256;                  // 512 float4 slots per tile
      int k = slot >> 5, w4 = (slot & 31) * 4;
      async_b128(vB + (unsigned)(k * LPAD + w4) * 4u, vh + k * W_ + w4);
      int w = slot >> 2, k4 = (slot & 3) * 4;
      async_b128(aB + (unsigned)(w * APAD + k4) * 4u, aw + w * 16 + k4);
    }
  }
  asm volatile("s_wait_asynccnt 0" ::: "memory");
  __syncthreads();

  int lm = lane & 15, koff = (lane >> 4) * 2;
  const float* aH = att  + (size_t)b * EROW + (wave * 16 + lm) * 16;   // attH row
  const float* aV = valW + ((size_t)bc * H_ + wave * 16 + lm) * KP_;   // valW row
  int mrow = wave * 16 + ((lane >> 4) << 3);

  for (int nt = 0; nt < 8; ++nt) {
    v8f acc = {};
#pragma unroll
    for (int s = 0; s < 4; ++s) {        // GEMM1: attH x valH
      int kk = s * 4 + koff;
      v2f a  = { aH[kk], aH[kk + 1] };
      v2f bb = { VHs[kk][nt * 16 + lm], VHs[kk + 1][nt * 16 + lm] };
      acc = __builtin_amdgcn_wmma_f32_16x16x4_f32(false, a, false, bb,
                                                  (short)0, acc, false, false);
    }
#pragma unroll
    for (int s = 0; s < 4; ++s) {        // GEMM2: valW x attW^T (accumulated)
      int kk = s * 4 + koff;
      v2f a  = { aV[kk], aV[kk + 1] };
      v2f bb = { AWs[nt * 16 + lm][kk], AWs[nt * 16 + lm][kk + 1] };
      acc = __builtin_amdgcn_wmma_f32_16x16x4_f32(false, a, false, bb,
                                                  (short)0, acc, false, false);
    }
#pragma unroll
    for (int j = 0; j < 8; ++j) {        // epilogue: 0.5*acc + residual
      int m = mrow + j;
      size_t o = ((size_t)bc * H_ + m) * W_ + nt * 16 + lm;
      out[o] = 0.5f * acc[j] + t2[o];
    }
  }
}

// ---------------------------------------------------------------------------
extern "C" void kernel_launch(void* const* d_in, const int* in_sizes, int n_in,
                              void* d_out, int out_size, void* d_ws, size_t ws_size,
                              hipStream_t stream) {
  const float* t1 = (const float*)d_in[0];
  const float* t2 = (const float*)d_in[1];
  float* out = (float*)d_out;
  float* ws  = (float*)d_ws;

  float* keyH   = ws + KH_OFF;
  float* keyW   = ws + KW_OFF;
  float* valH   = ws + VH_OFF;
  float* valW   = ws + VW_OFF;
  float* epart  = ws + ENP_OFF;
  float* energy = ws + EN_OFF;
  float* attb   = ws + ATT_OFF;
  float* mm     = ws + MM_OFF;

  pool_kernel<<<16384, 256, 0, stream>>>(t1, t2, keyH, keyW, valH, valW);
  energy_kernel<<<B_ * 16 * CSPLIT, 256, 0, stream>>>(t2, keyH, keyW, epart);
  reduce_energy_kernel<<<(B_ * EROW) / 256, 256, 0, stream>>>(epart, energy);
  minmax_kernel<<<1, 256, 0, stream>>>(energy, mm);
  softmax_kernel<<<B_, 256, 0, stream>>>(energy, mm, attb);
  out_kernel<<<B_ * C_, 256, 0, stream>>>(t2, valH, valW, attb, out);
}